// GCN_43868795961778
// MI455X (gfx1250) — compile-verified
//
#include <hip/hip_runtime.h>

#define NUM_USERS 100000
#define NUM_ITEMS 100000
#define N_NODES   200000
#define EMB_DIM   64
#define HID_DIM   128
#define NUM_EDGES 3200000

typedef float v2f __attribute__((ext_vector_type(2)));
typedef float v8f __attribute__((ext_vector_type(8)));

// ---------------------------------------------------------------------------
// gfx1250 async global -> LDS helpers (ASYNCcnt path)
//   LDS operand: wave-relative byte offset == low 32 bits of a flat pointer
//   to a __shared__ object (LDS aperture lives entirely in the high 32 bits).
// ---------------------------------------------------------------------------
__device__ __forceinline__ unsigned lds_offset(const void* p) {
    return (unsigned)(unsigned long long)p;
}
__device__ __forceinline__ void async_to_lds_b128(unsigned lds, const void* gptr) {
    asm volatile("global_load_async_to_lds_b128 %0, %1, off"
                 :: "v"(lds), "v"(gptr) : "memory");
}
__device__ __forceinline__ void wait_asynccnt0() {
    asm volatile("s_wait_asynccnt 0x0" ::: "memory");
}

// ---------------------------------------------------------------------------
// Utility: zero-fill (float4 granularity; all our sizes are /4)
// ---------------------------------------------------------------------------
__global__ void k_fill_zero(float* __restrict__ p, long n) {
    long i = ((long)blockIdx.x * 256 + threadIdx.x) * 4;
    if (i < n) *(float4*)(p + i) = make_float4(0.f, 0.f, 0.f, 0.f);
}

// ---------------------------------------------------------------------------
// Degree / symmetric normalization: deg = 1 (self loop) + count(dst); dinv=rsqrt
// ---------------------------------------------------------------------------
__global__ void k_deg_init(float* __restrict__ deg) {
    int i = blockIdx.x * 256 + threadIdx.x;
    if (i < N_NODES) deg[i] = 1.0f;
}
__global__ void k_deg_acc(const int* __restrict__ dst, float* __restrict__ deg) {
    long i = (long)blockIdx.x * 256 + threadIdx.x;
    if (i < NUM_EDGES) atomicAdd(&deg[dst[i]], 1.0f);
}
__global__ void k_deg_fin(float* __restrict__ deg) {
    int i = blockIdx.x * 256 + threadIdx.x;
    if (i < N_NODES) deg[i] = rsqrtf(deg[i]);
}

// ---------------------------------------------------------------------------
// WMMA fp32 GEMM:  H[M,N] = X[M,K] @ W[K,N]
//   - block = 256 threads = 8 waves (wave32)
//   - NT = N/16 column tiles, MT = 8/NT row tiles per block
//   - W and the A row-block staged in LDS via global_load_async_to_lds_b128
//     (ASYNCcnt), padded rows to kill ds bank conflicts
//   - each wave: K/4 x v_wmma_f32_16x16x4_f32 into a v8f accumulator
//   - X split across two source buffers at row `split` (user/item concat);
//     block row-tiles never straddle the split (100000 % 32 == 0)
// ---------------------------------------------------------------------------
template <int K, int N>
__global__ __launch_bounds__(256) void k_gemm_wmma(
    const float* __restrict__ X0, const float* __restrict__ X1, int split,
    const float* __restrict__ W, float* __restrict__ H)
{
    constexpr int NT  = N / 16;   // column tiles (waves across N)
    constexpr int MT  = 8 / NT;   // row tiles per block
    constexpr int LDA = K + 4;    // LDS row pad: lane stride 4 banks, conflict-free
    __shared__ __align__(16) float sW[K * N];
    __shared__ __align__(16) float sA[MT * 16 * LDA];

    const int tid = threadIdx.x;
    const unsigned sW_base = lds_offset(sW);
    const unsigned sA_base = lds_offset(sA);

    // stage W: K*N/4 float4 chunks, contiguous (multiple of 256 chunks)
    constexpr int WCHUNKS = K * N / 4;
    for (int i = tid; i < WCHUNKS; i += 256)
        async_to_lds_b128(sW_base + (unsigned)i * 16u, W + (size_t)i * 4);

    // stage A row-block: MT*16 rows of K floats -> padded rows in LDS
    const int m0 = blockIdx.x * (MT * 16);
    constexpr int ACHUNKS = MT * 16 * (K / 4);
    for (int i = tid; i < ACHUNKS; i += 256) {
        const int r  = i / (K / 4);
        const int c4 = (i - r * (K / 4)) * 4;
        const int row = m0 + r;
        const float* g = (row < split) ? X0 + (size_t)row * K + c4
                                       : X1 + (size_t)(row - split) * K + c4;
        async_to_lds_b128(sA_base + (unsigned)(r * LDA + c4) * 4u, g);
    }
    wait_asynccnt0();
    __syncthreads();

    const int wave  = tid >> 5;
    const int lane  = tid & 31;
    const int mt    = wave / NT;
    const int nt    = wave % NT;
    const int l16   = lane & 15;   // A: row in tile / B,D: column in tile
    const int khalf = lane >> 4;   // 0: K pair {0,1}, 1: K pair {2,3}

    v8f acc = {0.f, 0.f, 0.f, 0.f, 0.f, 0.f, 0.f, 0.f};

#pragma unroll
    for (int k0 = 0; k0 < K; k0 += 4) {
        v2f a, b;
        // A 16x4 f32 layout: lane l16 = M, VGPR v holds K = k0 + 2*khalf + v
        const float* ap = &sA[(mt * 16 + l16) * LDA + k0 + 2 * khalf];
        a[0] = ap[0];
        a[1] = ap[1];
        // B 4x16 f32 layout: lane l16 = N, VGPR v holds K = k0 + 2*khalf + v
        const float* bp = &sW[(k0 + 2 * khalf) * N + nt * 16 + l16];
        b[0] = bp[0];
        b[1] = bp[N];
        acc = __builtin_amdgcn_wmma_f32_16x16x4_f32(
            /*neg_a=*/false, a, /*neg_b=*/false, b,
            /*c_mod=*/(short)0, acc, /*reuse_a=*/false, /*reuse_b=*/false);
    }

    // D 16x16 f32 layout: VGPR r -> M = r + 8*khalf, lane l16 -> N
    float* out = H + (size_t)(m0 + mt * 16 + 8 * khalf) * N + nt * 16 + l16;
#pragma unroll
    for (int r = 0; r < 8; ++r) out[(size_t)r * N] = acc[r];
}

// ---------------------------------------------------------------------------
// Edge scatter:  AGG[dst] += H[src] * dinv[src]*dinv[dst]
//   F/4 threads per edge, float4 gather, 4 f32 atomics (resolve in L2:
//   H and AGG both fit in the 192MB L2).
// ---------------------------------------------------------------------------
template <int F>
__global__ __launch_bounds__(256) void k_scatter(
    const int* __restrict__ src, const int* __restrict__ dst,
    const float* __restrict__ dinv, const float* __restrict__ H,
    float* __restrict__ AGG, int nE)
{
    constexpr int TPE = F / 4;
    const long gid = (long)blockIdx.x * 256 + threadIdx.x;
    const long e   = gid / TPE;
    if (e >= nE) return;
    const int c = (int)(gid % TPE) * 4;

    const int s = src[e];
    const int d = dst[e];
    const float nrm = dinv[s] * dinv[d];

    const float4 h = *(const float4*)(H + (size_t)s * F + c);
    float* o = AGG + (size_t)d * F + c;
    atomicAdd(o + 0, h.x * nrm);
    atomicAdd(o + 1, h.y * nrm);
    atomicAdd(o + 2, h.z * nrm);
    atomicAdd(o + 3, h.w * nrm);
}

// ---------------------------------------------------------------------------
// Epilogue: agg[i,f] = (agg + h*dinv^2 (self loop) + b[f]) , optional ReLU
// ---------------------------------------------------------------------------
template <int F, bool RELU>
__global__ __launch_bounds__(256) void k_finish(
    float* __restrict__ agg, const float* __restrict__ h,
    const float* __restrict__ dinv, const float* __restrict__ b)
{
    const long idx = (long)blockIdx.x * 256 + threadIdx.x;
    if (idx >= (long)N_NODES * F) return;
    const int node = (int)(idx / F);
    const int f    = (int)(idx % F);
    const float di = dinv[node];
    float v = agg[idx] + h[idx] * di * di + b[f];
    agg[idx] = RELU ? fmaxf(v, 0.0f) : v;
}

// ---------------------------------------------------------------------------
// Orchestration
// ---------------------------------------------------------------------------
extern "C" void kernel_launch(void* const* d_in, const int* in_sizes, int n_in,
                              void* d_out, int out_size, void* d_ws, size_t ws_size,
                              hipStream_t stream)
{
    const int*   edge = (const int*)d_in[0];        // [2, E] row-major
    const int*   src  = edge;                       // edge_index[0]
    const int*   dst  = edge + NUM_EDGES;           // edge_index[1]
    const float* uemb = (const float*)d_in[1];      // [100000, 64]
    const float* iemb = (const float*)d_in[2];      // [100000, 64]
    const float* W1   = (const float*)d_in[3];      // [64, 128]
    const float* b1   = (const float*)d_in[4];      // [128]
    const float* W2   = (const float*)d_in[5];      // [128, 64]
    const float* b2   = (const float*)d_in[6];      // [64]
    float*       out  = (float*)d_out;              // [200000, 64] flat

    // workspace layout (~206 MB)
    char*  ws   = (char*)d_ws;
    float* dinv = (float*)ws;                                   // 0.8 MB
    float* h1   = (float*)(ws + (size_t)(1u << 20));            // 102.4 MB (reused for h2)
    float* agg1 = (float*)(ws + (size_t)(1u << 20) + 102400000);// 102.4 MB (becomes x1)

    const int B = 256;

    // 1) symmetric-normalized degrees
    k_deg_init<<<(N_NODES + B - 1) / B, B, 0, stream>>>(dinv);
    k_deg_acc<<<(NUM_EDGES + B - 1) / B, B, 0, stream>>>(dst, dinv);
    k_deg_fin<<<(N_NODES + B - 1) / B, B, 0, stream>>>(dinv);

    // 2) zero aggregation buffers (agg1 + final output used as agg2)
    {
        long n1 = (long)N_NODES * HID_DIM;
        long n2 = (long)N_NODES * EMB_DIM;
        k_fill_zero<<<(unsigned)((n1 / 4 + B - 1) / B), B, 0, stream>>>(agg1, n1);
        k_fill_zero<<<(unsigned)((n2 / 4 + B - 1) / B), B, 0, stream>>>(out, n2);
    }

    // 3) layer 1: h1 = concat(user,item) @ W1   (WMMA fp32)
    k_gemm_wmma<EMB_DIM, HID_DIM><<<N_NODES / 16, B, 0, stream>>>(
        uemb, iemb, NUM_USERS, W1, h1);

    // 4) layer 1 edge aggregation
    {
        long t = (long)NUM_EDGES * (HID_DIM / 4);
        k_scatter<HID_DIM><<<(unsigned)((t + B - 1) / B), B, 0, stream>>>(
            src, dst, dinv, h1, agg1, NUM_EDGES);
    }

    // 5) self-loop + bias + relu  ->  x1 (in place in agg1)
    {
        long t = (long)N_NODES * HID_DIM;
        k_finish<HID_DIM, true><<<(unsigned)((t + B - 1) / B), B, 0, stream>>>(
            agg1, h1, dinv, b1);
    }

    // 6) layer 2: h2 = x1 @ W2   (WMMA fp32; h2 reuses h1 buffer)
    float* h2 = h1;
    k_gemm_wmma<HID_DIM, EMB_DIM><<<N_NODES / 32, B, 0, stream>>>(
        agg1, agg1, N_NODES, W2, h2);

    // 7) layer 2 edge aggregation directly into d_out
    {
        long t = (long)NUM_EDGES * (EMB_DIM / 4);
        k_scatter<EMB_DIM><<<(unsigned)((t + B - 1) / B), B, 0, stream>>>(
            src, dst, dinv, h2, out, NUM_EDGES);
    }

    // 8) self-loop + bias (no relu), in place in d_out
    {
        long t = (long)N_NODES * EMB_DIM;
        k_finish<EMB_DIM, false><<<(unsigned)((t + B - 1) / B), B, 0, stream>>>(
            out, h2, dinv, b2);
    }
}